// Attention_32049045963483
// MI455X (gfx1250) — compile-verified
//
#include <hip/hip_runtime.h>
#include <hip/hip_bf16.h>
#include <math.h>

// ---------------------------------------------------------------------------
// Fused low-res attention for MI455X (gfx1250, wave32).
//
// Key algebraic collapse: the x64 nearest-neighbor upsample of the 64x64
// score map means softmax/attn only ever see 64 distinct key scores per
// query-row, and the output is constant along y.  We never materialize the
// [8,4096,4096] attention tensor (~1GB); total work is ~230 MFLOP over a
// ~6MB working set -> latency bound, everything L2-resident.
//
// fp32 matrix math runs on V_WMMA_F32_16X16X4_F32 (CDNA5 fp32 WMMA), so we
// keep full reference precision while using the matrix pipes.  The per-head
// score GEMM stages its operands into LDS with GLOBAL_LOAD_ASYNC_TO_LDS
// (ASYNCcnt-tracked) before feeding the WMMA fragments.
// ---------------------------------------------------------------------------

typedef __attribute__((ext_vector_type(2))) float v2f;
typedef __attribute__((ext_vector_type(8))) float v8f;

// D = A(16x4 f32) * B(4x16 f32) + C(16x16 f32), wave32.
// A layout (ISA 7.12.2): lanes 0-15 hold M=lane {v0:K0, v1:K1};
//                        lanes 16-31 hold M=lane-16 {v0:K2, v1:K3}.
// B layout (row-striped): v0 lanes 0-15 = row K0, lanes 16-31 = row K2;
//                         v1 = rows K1 / K3.  N = lane & 15.
// C/D: VGPR r, lanes 0-15 -> M=r, lanes 16-31 -> M=r+8; N = lane & 15.
static __device__ __forceinline__ v8f wmma4(v2f a, v2f b, v8f c) {
  return __builtin_amdgcn_wmma_f32_16x16x4_f32(
      /*neg_a=*/false, a, /*neg_b=*/false, b,
      /*c_mod=*/(short)0, c, /*reuse_a=*/false, /*reuse_b=*/false);
}

// ---------------------------------------------------------------------------
// Kernel 1: qkv 1x1-conv == GEMM  C[192,4096] = W[192,64] * F[64,4096]
// One 16x16 tile per wave, 3072 tiles, K-loop 16 x (16x16x4 WMMA).
// ---------------------------------------------------------------------------
__global__ void qkv_gemm(const float* __restrict__ W, const float* __restrict__ F,
                         float* __restrict__ out) {
  const int lane = threadIdx.x & 31;
  const int wave = threadIdx.x >> 5;
  const int tile = blockIdx.x * 4 + wave;      // 0..3071
  const int tm = tile >> 8;                    // /256 -> 0..11
  const int tn = tile & 255;                   // 0..255
  const int l15 = lane & 15;
  const int half = lane >> 4;
  const int m = tm * 16 + l15;                 // output channel
  const int n = tn * 16 + l15;                 // spatial position

  v8f c = {};
#pragma unroll
  for (int k0 = 0; k0 < 64; k0 += 4) {
    const int kci = k0 + 2 * half;
    v2f a, b;
    a.x = W[m * 64 + kci];
    a.y = W[m * 64 + kci + 1];
    b.x = F[kci * 4096 + n];
    b.y = F[(kci + 1) * 4096 + n];
    c = wmma4(a, b, c);
  }
#pragma unroll
  for (int r = 0; r < 8; ++r)
    out[(tm * 16 + r + 8 * half) * 4096 + tn * 16 + l15] = c[r];
}

// ---------------------------------------------------------------------------
// Kernel 2a: weight pre-transpose  wT[qk][tap][ci][oc] = w[oc][ci][tap]
// so the dconv A-fragment loads are lane-coalesced (64B per half-wave).
// One-shot 4MB stream, coalesced writes, deterministic.
// ---------------------------------------------------------------------------
__global__ void wtrans(const float* __restrict__ wq, const float* __restrict__ wk,
                       float* __restrict__ wT) {
  const int qk = blockIdx.z;
  const int o = blockIdx.x * blockDim.x + threadIdx.x;  // 0..495615
  const int tap = o >> 12;
  const int ci = (o >> 6) & 63;
  const int oc = o & 63;
  const float* __restrict__ w = qk ? wk : wq;
  wT[qk * 495616 + o] = w[(oc * 64 + ci) * 121 + tap];
}

// ---------------------------------------------------------------------------
// Kernel 2b: 11x11 stride-8 pad-2 conv as 121 accumulated tap-GEMMs
// (64oc x 64pos x 64ic per tap).  Split-K across kx (11-way) into
// deterministic per-(kx,qk) partial buffers -- no float atomics, so the
// result is bit-deterministic across replays.
// grid (4, 11, 2) x 128 threads:   wave -> one 16x16 tile of (oc,pos).
// ---------------------------------------------------------------------------
__global__ void dconv_part(const float* __restrict__ qkv,
                           const float* __restrict__ wT,
                           float* __restrict__ part) {
  const int lane = threadIdx.x & 31;
  const int wave = threadIdx.x >> 5;
  const int tile = blockIdx.x * 4 + wave;      // 0..15
  const int tm = tile >> 2, tn = tile & 3;
  const int kx = blockIdx.y;                   // 0..10
  const int qk = blockIdx.z;                   // 0:q 1:k
  const int l15 = lane & 15;
  const int half = lane >> 4;

  const float* __restrict__ src = qkv + qk * 64 * 4096;  // q or k feature map

  const int m = tm * 16 + l15;                 // out channel
  const int pos = tn * 16 + l15;               // xl*8+yl, low-res position
  const int xl = pos >> 3, yl = pos & 7;
  const int x = xl * 8 - 2 + kx;               // input row for this kx
  const bool xv = (x >= 0) && (x < 64);

  v8f c = {};
  for (int ky = 0; ky < 11; ++ky) {
    const int y = yl * 8 - 2 + ky;
    const bool valid = xv && (y >= 0) && (y < 64);
    const int sidx = valid ? (x * 64 + y) : 0; // safe dummy when padded
    const int tap = kx * 11 + ky;
    // wbase[ci][oc]: lanes 0-15 read 64 consecutive bytes -> coalesced.
    const float* __restrict__ wbase = wT + ((size_t)(qk * 121 + tap)) * 4096;
    // prefetch next tap's weight line (global_prefetch_b8)
    if (ky < 10) __builtin_prefetch(wbase + 4096 + m, 0, 0);
#pragma unroll
    for (int k0 = 0; k0 < 64; k0 += 4) {
      const int kci = k0 + 2 * half;
      v2f a, b;
      a.x = wbase[kci * 64 + m];
      a.y = wbase[(kci + 1) * 64 + m];
      b.x = valid ? src[kci * 4096 + sidx] : 0.0f;
      b.y = valid ? src[(kci + 1) * 4096 + sidx] : 0.0f;
      c = wmma4(a, b, c);
    }
  }
  float* __restrict__ dst = part + (kx * 2 + qk) * 4096;  // [64oc x 64pos]
#pragma unroll
  for (int r = 0; r < 8; ++r)
    dst[(tm * 16 + r + 8 * half) * 64 + tn * 16 + l15] = c[r];
}

// ---------------------------------------------------------------------------
// Kernel 3: reduce the 11 kx-partials, + bias, exact GELU, q-scale.
// ---------------------------------------------------------------------------
__global__ void reduce_bias_gelu(const float* __restrict__ part,
                                 const float* __restrict__ bq,
                                 const float* __restrict__ bk,
                                 float* __restrict__ qlow,
                                 float* __restrict__ klow) {
  const int t = blockIdx.x * blockDim.x + threadIdx.x;  // 0..8191
  const int qk = t >> 12;
  const int mn = t & 4095;
  const int m = mn >> 6;
  float s = 0.0f;
#pragma unroll
  for (int kx = 0; kx < 11; ++kx) s += part[(kx * 2 + qk) * 4096 + mn];
  s += qk ? bk[m] : bq[m];
  // exact GELU: x * 0.5 * (1 + erf(x / sqrt(2)))
  const float g = 0.5f * s * (1.0f + erff(s * 0.70710678118654752f));
  if (qk == 0)
    qlow[mn] = g * 0.35355339059327373f;  // * DIM_HEAD^-0.5
  else
    klow[mn] = g;
}

// ---------------------------------------------------------------------------
// Kernel 4: vbar[ch, row] = mean over y of v[ch, row, y]   (v = qkv rows 128+)
// ---------------------------------------------------------------------------
__global__ void vbar_kernel(const float* __restrict__ qkv, float* __restrict__ vbar) {
  const int t = blockIdx.x * blockDim.x + threadIdx.x;  // 0..4095
  const int ch = t >> 6, row = t & 63;
  const float* __restrict__ p = qkv + (128 + ch) * 4096 + row * 64;
  float s = 0.0f;
#pragma unroll 8
  for (int y = 0; y < 64; ++y) s += p[y];
  vbar[t] = s * (1.0f / 64.0f);
}

// ---------------------------------------------------------------------------
// Kernel 5 (per head): async-stage qlow/klow head slices (2KB each) into LDS
// (GLOBAL_LOAD_ASYNC_TO_LDS_B128, drained with s_wait_asynccnt), then
// dots = qlow^T * klow (WMMA, K=8), softmax over the 64 low-res keys in LDS,
// then out[ch, x, :] = sum_jl p[x,jl]*vbar[ch,jl] broadcast across y with
// float4 stores.  One block of 4 waves per head.
// ---------------------------------------------------------------------------
__global__ void attn_out(const float* __restrict__ qlow,
                         const float* __restrict__ klow,
                         const float* __restrict__ vbar,
                         float* __restrict__ out) {
  __shared__ float sd[64][65];                 // +1 pad: conflict-free rows
  __shared__ float sq[512];                    // head slice of qlow [8ci][64pos]
  __shared__ float sk[512];                    // head slice of klow
  const int h = blockIdx.x;                    // head
  const int lane = threadIdx.x & 31;
  const int wave = threadIdx.x >> 5;           // 0..3 == tile row tm
  const int l15 = lane & 15;
  const int half = lane >> 4;

  // Async copy: head slices are contiguous (512 floats each).  128 threads x
  // two B128 apiece (one per slice) covers both slices.  ASYNCcnt-tracked; no VGPR staging.
  {
    const float* gq = qlow + h * 512 + threadIdx.x * 4;
    const float* gk = klow + h * 512 + threadIdx.x * 4;
    unsigned lq = (unsigned)(size_t)(&sq[threadIdx.x * 4]);
    unsigned lk = (unsigned)(size_t)(&sk[threadIdx.x * 4]);
    asm volatile("global_load_async_to_lds_b128 %0, %1, off"
                 :: "v"(lq), "v"(gq) : "memory");
    asm volatile("global_load_async_to_lds_b128 %0, %1, off"
                 :: "v"(lk), "v"(gk) : "memory");
    asm volatile("s_wait_asynccnt 0" ::: "memory");
  }
  __syncthreads();

  // dots tiles: wave w handles tiles (tm=w, tn=0..3); A[m=i_low][k=c] reads
  // the q slice transposed, B[k=c][n=j_low] reads the k slice directly.
  for (int tn = 0; tn < 4; ++tn) {
    v8f c = {};
#pragma unroll
    for (int k0 = 0; k0 < 8; k0 += 4) {
      const int kci = k0 + 2 * half;
      v2f a, b;
      a.x = sq[kci * 64 + wave * 16 + l15];
      a.y = sq[(kci + 1) * 64 + wave * 16 + l15];
      b.x = sk[kci * 64 + tn * 16 + l15];
      b.y = sk[(kci + 1) * 64 + tn * 16 + l15];
      c = wmma4(a, b, c);
    }
#pragma unroll
    for (int r = 0; r < 8; ++r)
      sd[wave * 16 + r + 8 * half][tn * 16 + l15] = c[r];
  }
  __syncthreads();

  // Row softmax over the 64 unique key scores (upsample makes the 4096-wide
  // softmax equal to this divided by 64; the /64 folds into vbar's mean).
  if (threadIdx.x < 64) {
    const int i = threadIdx.x;
    float mx = -INFINITY;
    for (int j = 0; j < 64; ++j) mx = fmaxf(mx, sd[i][j]);
    float ssum = 0.0f;
    for (int j = 0; j < 64; ++j) {
      const float e = expf(sd[i][j] - mx);
      sd[i][j] = e;
      ssum += e;
    }
    const float inv = 1.0f / ssum;
    for (int j = 0; j < 64; ++j) sd[i][j] *= inv;
  }
  __syncthreads();

  // out[h*8+cl][x][y] = sum_jl p[x][jl] * vbar[h*8+cl][jl], constant in y.
  for (int i = 0; i < 4; ++i) {
    const int idx = i * 128 + threadIdx.x;     // 0..511 = (cl, x)
    const int cl = idx >> 6;
    const int x = idx & 63;
    const float* __restrict__ vb = vbar + (h * 8 + cl) * 64;
    float s = 0.0f;
#pragma unroll 8
    for (int j = 0; j < 64; ++j) s += sd[x][j] * vb[j];
    float4 v4 = make_float4(s, s, s, s);
    float4* __restrict__ dst = (float4*)(out + (h * 8 + cl) * 4096 + x * 64);
#pragma unroll
    for (int w = 0; w < 16; ++w) dst[w] = v4;
  }
}

// ---------------------------------------------------------------------------
// Workspace layout (floats):
//   qkv   : [192 x 4096]          @ 0        (786432)
//   part  : [11*2 x 64 x 64]      @ 786432   (90112)
//   qlow  : [64 x 64]             @ 876544   (4096)
//   klow  : [64 x 64]             @ 880640   (4096)
//   vbar  : [64 x 64]             @ 884736   (4096)
//   wT    : [2 x 121 x 64 x 64]   @ 888832   (991232)
// total ~7.5 MB
// ---------------------------------------------------------------------------
extern "C" void kernel_launch(void* const* d_in, const int* in_sizes, int n_in,
                              void* d_out, int out_size, void* d_ws, size_t ws_size,
                              hipStream_t stream) {
  const float* f     = (const float*)d_in[0];  // [1,64,64,64]
  const float* w_qkv = (const float*)d_in[1];  // [192,64,1,1]
  const float* wq    = (const float*)d_in[2];  // [64,64,11,11]
  const float* bq    = (const float*)d_in[3];  // [64]
  const float* wk    = (const float*)d_in[4];  // [64,64,11,11]
  const float* bk    = (const float*)d_in[5];  // [64]
  float* out = (float*)d_out;                  // [1,64,64,64]

  float* ws   = (float*)d_ws;
  float* qkv  = ws;
  float* part = ws + 786432;
  float* qlow = ws + 876544;
  float* klow = ws + 880640;
  float* vbar = ws + 884736;
  float* wT   = ws + 888832;

  qkv_gemm<<<768, 128, 0, stream>>>(w_qkv, f, qkv);
  wtrans<<<dim3(1936, 1, 2), 256, 0, stream>>>(wq, wk, wT);
  dconv_part<<<dim3(4, 11, 2), 128, 0, stream>>>(qkv, wT, part);
  reduce_bias_gelu<<<32, 256, 0, stream>>>(part, bq, bk, qlow, klow);
  vbar_kernel<<<16, 256, 0, stream>>>(qkv, vbar);
  attn_out<<<8, 128, 0, stream>>>(qlow, klow, vbar, out);
}